// GraphConvolution_23648089932274
// MI455X (gfx1250) — compile-verified
//
#include <hip/hip_runtime.h>

typedef float v2f __attribute__((ext_vector_type(2)));
typedef float v8f __attribute__((ext_vector_type(8)));

#define DIN  256
#define DOUT 128
#define KDEG 16

// ---------------------------------------------------------------------------
// Stage 1: H = relu(feats @ W + b)
//   feats: [n, 256] row-major f32; W: [256, 128]; b: [128]; H: [n, 128] (ws)
//
// Block = 256 threads = 8 wave32s. Block tile = 128 rows x 128 cols.
// Wave w computes rows [blockRow + 16w, +16) x all 128 cols via
// V_WMMA_F32_16X16X4_F32.
//
// W is staged into LDS in two 64KB K-halves, K-PAIR-INTERLEAVED:
//   sW[p*256 + 2*c + e] = W[kbase + 2p + e][c]   (p = k-pair, c = col)
// so a lane's B fragment {W[k][c], W[k+1][c]} is one contiguous 8B chunk
// -> single ds_load_b64 straight into the WMMA operand pair, and the 8
// column tiles are immediate offsets (t*128B) off one address register.
//
// A fragments are double-buffered in 32-k chunks (8 global_load_b64 issued
// for chunk ch+1 while chunk ch feeds 64 WMMAs).
// ---------------------------------------------------------------------------
__global__ __launch_bounds__(256) void gemm_bias_relu_f32wmma(
    const float* __restrict__ A,     // feats
    const float* __restrict__ W,     // weights
    const float* __restrict__ bias,  // b
    float* __restrict__ H,           // output h
    int n)
{
    __shared__ float sW[128 * 128];  // one K-half of W, pair-interleaved: 64KB

    const int tid  = threadIdx.x;
    const int lane = tid & 31;
    const int wave = tid >> 5;
    const int hi   = lane >> 4;      // lane half: 0 or 1
    const int nlo  = lane & 15;      // col-within-tile / M-within-tile
    const int klo  = hi * 2;         // lane's base K inside a 4-step

    const int rowBase = blockIdx.x * 128 + wave * 16;

    // Clamped row for A loads (tail rows recompute row n-1; stores predicated)
    int mRow = rowBase + nlo;
    if (mRow > n - 1) mRow = n - 1;
    if (mRow < 0) mRow = 0;

    v8f acc[8];
#pragma unroll
    for (int t = 0; t < 8; ++t) acc[t] = v8f{};

    for (int halfk = 0; halfk < 2; ++halfk) {
        // ---- cooperative pair-interleaved stage of W-half into LDS ----
        const float* wsrc = W + halfk * (128 * 128);
#pragma unroll
        for (int j = 0; j < 8; ++j) {
            const int id = tid + j * 256;        // 0..2047 tasks
            const int p  = id >> 5;              // k-pair 0..63
            const int c  = (id & 31) * 4;        // col quad, coalesced rows
            const float4 r0 = *(const float4*)(wsrc + (2 * p)     * 128 + c);
            const float4 r1 = *(const float4*)(wsrc + (2 * p + 1) * 128 + c);
            v2f* dst = (v2f*)(&sW[p * 256 + c * 2]);
            v2f t0; t0.x = r0.x; t0.y = r1.x;
            v2f t1; t1.x = r0.y; t1.y = r1.y;
            v2f t2; t2.x = r0.z; t2.y = r1.z;
            v2f t3; t3.x = r0.w; t3.y = r1.w;
            dst[0] = t0; dst[1] = t1; dst[2] = t2; dst[3] = t3;
        }
        __syncthreads();

        // per-lane A stream: A[mRow][halfk*128 + klo + kk .. +1]
        const float* aPtr = A + (size_t)mRow * DIN + halfk * 128 + klo;

        // ---- A double-buffer: 4 chunks of 8 k-steps (32 k-values) ----
        v2f aBuf[2][8];
#pragma unroll
        for (int j = 0; j < 8; ++j)
            aBuf[0][j] = *(const v2f*)(aPtr + j * 4);

#pragma unroll
        for (int ch = 0; ch < 4; ++ch) {
            const int cur = ch & 1;
            const int nxt = cur ^ 1;
            if (ch < 3) {
#pragma unroll
                for (int j = 0; j < 8; ++j)
                    aBuf[nxt][j] = *(const v2f*)(aPtr + (ch + 1) * 32 + j * 4);
            }
#pragma unroll
            for (int j = 0; j < 8; ++j) {
                const int kk = ch * 32 + j * 4;
                // pair index p = kk/2 + hi; float base = p*256 + nlo*2
                const int pb = (kk / 2 + hi) * 256 + nlo * 2;
#pragma unroll
                for (int t = 0; t < 8; ++t) {
                    const v2f bb = *(const v2f*)(&sW[pb + t * 32]); // ds_load_b64
                    acc[t] = __builtin_amdgcn_wmma_f32_16x16x4_f32(
                        false, aBuf[cur][j], false, bb, (short)0,
                        acc[t], false, false);
                }
            }
        }
        __syncthreads();
    }

    // ---- epilogue: bias + relu + predicated store ----
    // C/D layout: VGPR r of lane L holds element (M = r + 8*(L/16), N = L%16)
#pragma unroll
    for (int t = 0; t < 8; ++t) {
        const float bv = bias[t * 16 + nlo];
#pragma unroll
        for (int r = 0; r < 8; ++r) {
            const int m = rowBase + r + hi * 8;
            if (m < n) {
                float v = acc[t][r] + bv;
                H[(size_t)m * DOUT + t * 16 + nlo] = v > 0.0f ? v : 0.0f;
            }
        }
    }
}

// ---------------------------------------------------------------------------
// Stage 2: out[i] = mean_k H[edge[i][k]]
// One wave per node; each lane owns 4 consecutive floats (float4).
// Neighbor rows are 512B contiguous -> coalesced b128 loads, serviced by L2
// (H is 51.2 MB < 192 MB L2). Indices are wave-uniform -> scalar loads.
// ---------------------------------------------------------------------------
__global__ __launch_bounds__(256) void gather_mean(
    const float* __restrict__ H,
    const long long* __restrict__ edges,  // [n, 16] int64
    float* __restrict__ out,
    int n)
{
    const int node = blockIdx.x * 8 + (threadIdx.x >> 5);
    const int lane = threadIdx.x & 31;
    if (node >= n) return;

    const long long* ep = edges + (size_t)node * KDEG;

    float4 acc = make_float4(0.f, 0.f, 0.f, 0.f);
#pragma unroll
    for (int k = 0; k < KDEG; ++k) {
        const long long idx = ep[k];
        const float4 v = *(const float4*)(H + (size_t)idx * DOUT + lane * 4);
        acc.x += v.x; acc.y += v.y; acc.z += v.z; acc.w += v.w;
    }

    const float s = 1.0f / (float)KDEG;
    float4 r = make_float4(acc.x * s, acc.y * s, acc.z * s, acc.w * s);
    *(float4*)(out + (size_t)node * DOUT + lane * 4) = r;
}

extern "C" void kernel_launch(void* const* d_in, const int* in_sizes, int n_in,
                              void* d_out, int out_size, void* d_ws, size_t ws_size,
                              hipStream_t stream) {
    const float*     feats = (const float*)d_in[0];      // [n, 256]
    const long long* edges = (const long long*)d_in[1];  // [n, 16] int64
    const float*     W     = (const float*)d_in[2];      // [256, 128]
    const float*     bias  = (const float*)d_in[3];      // [128]
    float* out = (float*)d_out;                          // [n, 128]
    float* h   = (float*)d_ws;                           // [n, 128] scratch

    const int n = in_sizes[0] / DIN;

    const int gridGemm = (n + 127) / 128;
    gemm_bias_relu_f32wmma<<<dim3(gridGemm), dim3(256), 0, stream>>>(
        feats, W, bias, h, n);

    const int gridAgg = (n + 7) / 8;
    gather_mean<<<dim3(gridAgg), dim3(256), 0, stream>>>(h, edges, out, n);
}